// AttentionBase_62483184222474
// MI455X (gfx1250) — compile-verified
//
#include <hip/hip_runtime.h>

typedef __attribute__((ext_vector_type(2))) float v2f;
typedef __attribute__((ext_vector_type(8))) float v8f;

#define SCOPE    5
#define T_DIM    2048
#define F_DIM    256
#define MASK_MAG 9999.0f
#define NTHREADS 256

__global__ __launch_bounds__(NTHREADS)
void attn_scope_kernel(const float* __restrict__ X,     // [B, T, F]
                       const int*   __restrict__ seq,   // [B]
                       const float* __restrict__ W,     // [SCOPE, F]
                       const float* __restrict__ bias,  // [1]
                       float* __restrict__ outResult,   // [B, F]
                       float* __restrict__ outWeight)   // [B, T]
{
    // wS16: W zero-padded to 16 rows (phase 1 only). Reused as probS[2048]
    // (8 KB out of 16 KB) in phases 2-3 -- the phases are disjoint.
    __shared__ float wS16[16][F_DIM];       // 16 KB
    __shared__ float yS[T_DIM + 16][SCOPE]; // ~41 KB : y[t][v] = X[t].W[v]
    __shared__ float red[16];

    float* const probS = &wS16[0][0];       // alias, phases 2-3

    const int b       = blockIdx.x;
    const int tid     = threadIdx.x;
    const int lane    = tid & 31;
    const int wave    = tid >> 5;           // 0..7
    const int lane16  = lane & 15;
    const int halfSel = lane >> 4;          // 0: lanes 0-15, 1: lanes 16-31
    const float* __restrict__ Xb = X + (size_t)b * T_DIM * F_DIM;

    // Stage W into LDS, rows [SCOPE,16) zeroed; zero y padding rows [T, T+16).
    for (int i = tid; i < 16 * F_DIM; i += NTHREADS) {
        const int r = i / F_DIM, c = i % F_DIM;
        wS16[r][c] = (r < SCOPE) ? W[r * F_DIM + c] : 0.0f;
    }
    for (int i = tid; i < 16 * SCOPE; i += NTHREADS)
        yS[T_DIM + i / SCOPE][i % SCOPE] = 0.0f;
    __syncthreads();

    // ---------- Phase 1: y[t][n] = X[t] . W[n] via V_WMMA_F32_16X16X4_F32 ----------
    // A (16x4): lane<16 -> row t0+lane16, K={0,1}; lane>=16 -> K={2,3}.
    // B (4x16): B[k][n] = Wpad[n][k] (rows n>=SCOPE are zero -> no divergence).
    // D (16x16): vgpr j: lane<16 -> M=j, lane>=16 -> M=j+8; N=lane16.
    for (int tile = wave; tile < T_DIM / 16; tile += NTHREADS / 32) {
        const int t0 = tile * 16;
        const float* __restrict__ rowA =
            Xb + (size_t)(t0 + lane16) * F_DIM + 2 * halfSel;
        const float* __restrict__ rowB = &wS16[lane16][2 * halfSel];
        v8f acc = {};
#pragma unroll 8
        for (int k0 = 0; k0 < F_DIM; k0 += 4) {
            v2f a  = *(const v2f*)(rowA + k0);   // global_load_b64
            v2f bf = *(const v2f*)(rowB + k0);   // ds_load_b64 (uniform EXEC)
            acc = __builtin_amdgcn_wmma_f32_16x16x4_f32(
                /*neg_a=*/false, a, /*neg_b=*/false, bf,
                /*c_mod=*/(short)0, acc, /*reuse_a=*/false, /*reuse_b=*/false);
        }
        if (lane16 < SCOPE) {
#pragma unroll
            for (int j = 0; j < 8; ++j)
                yS[t0 + halfSel * 8 + j][lane16] = acc[j];
        }
    }
    __syncthreads();

    // ---------- Phase 2: masked scores + softmax over T ----------
    const int   len = seq[b];
    const float bv  = bias[0];

    float lmax = -3.0e38f;
    for (int t = tid; t < T_DIM; t += NTHREADS) {
        float s = bv;
#pragma unroll
        for (int v = 0; v < SCOPE; ++v) s += yS[t + v][v];
        const float m = (t < len) ? MASK_MAG : -MASK_MAG;
        s = fminf(s, m);
        probS[t] = s;
        lmax = fmaxf(lmax, s);
    }
#pragma unroll
    for (int off = 16; off > 0; off >>= 1)
        lmax = fmaxf(lmax, __shfl_down(lmax, off, 32));
    if (lane == 0) red[wave] = lmax;
    __syncthreads();
    if (tid == 0) {
        float m = red[0];
        for (int w = 1; w < 8; ++w) m = fmaxf(m, red[w]);
        red[0] = m;
    }
    __syncthreads();
    const float gmax = red[0];

    float lsum = 0.0f;
    for (int t = tid; t < T_DIM; t += NTHREADS) {
        const float e = __expf(probS[t] - gmax);
        probS[t] = e;
        lsum += e;
    }
#pragma unroll
    for (int off = 16; off > 0; off >>= 1)
        lsum += __shfl_down(lsum, off, 32);
    if (lane == 0) red[8 + wave] = lsum;
    __syncthreads();
    if (tid == 0) {
        float s = 0.0f;
        for (int w = 0; w < 8; ++w) s += red[8 + w];
        red[8] = 1.0f / s;
    }
    __syncthreads();
    const float inv = red[8];
    for (int t = tid; t < T_DIM; t += NTHREADS) {
        const float p = probS[t] * inv;
        probS[t] = p;
        outWeight[(size_t)b * T_DIM + t] = p;
    }
    __syncthreads();

    // ---------- Phase 3: result[b][f] = sum_t p[t] * X[t][f] ----------
    const int f = tid;  // NTHREADS == F_DIM
    float a0 = 0.0f, a1 = 0.0f, a2 = 0.0f, a3 = 0.0f;
    for (int t = 0; t < T_DIM; t += 4) {
        a0 = fmaf(Xb[(size_t)(t + 0) * F_DIM + f], probS[t + 0], a0);
        a1 = fmaf(Xb[(size_t)(t + 1) * F_DIM + f], probS[t + 1], a1);
        a2 = fmaf(Xb[(size_t)(t + 2) * F_DIM + f], probS[t + 2], a2);
        a3 = fmaf(Xb[(size_t)(t + 3) * F_DIM + f], probS[t + 3], a3);
    }
    outResult[(size_t)b * F_DIM + f] = (a0 + a1) + (a2 + a3);
}

extern "C" void kernel_launch(void* const* d_in, const int* in_sizes, int n_in,
                              void* d_out, int out_size, void* d_ws, size_t ws_size,
                              hipStream_t stream) {
    // Inputs (setup_inputs order): dataInput f32 [32,2048,256], seqInput i32 [32],
    // W f32 [5,256], b f32 [1], hiddenNoduleNumbers i32 [1] (unused).
    const float* X    = (const float*)d_in[0];
    const int*   seq  = (const int*)  d_in[1];
    const float* W    = (const float*)d_in[2];
    const float* bias = (const float*)d_in[3];
    (void)in_sizes; (void)n_in; (void)d_ws; (void)ws_size; (void)out_size;

    float* out       = (float*)d_out;
    float* outResult = out;                 // [32, 256]
    float* outWeight = out + 32 * F_DIM;    // [32, 2048]

    attn_scope_kernel<<<32, NTHREADS, 0, stream>>>(X, seq, W, bias,
                                                   outResult, outWeight);
}